// MeanAggregator_90271622627847
// MI455X (gfx1250) — compile-verified
//
#include <hip/hip_runtime.h>
#include <hip/hip_bf16.h>

// GNN mean-aggregator for MI455X (gfx1250, wave32).
// Outputs (concat, each [B, D] f32):
//   0: mean_k features[neigh_idx[b][k]]
//   1: mean_k features[perm[neigh_idx[b][k]]]
//   2: features[nodes[b]]
//   3: features[perm[nodes[b]]]

#define D_FEAT     256
#define K_S        10
#define TILE_B     16
#define MEAN_THREADS 512              // 16 waves: wave w stages row w and owns feature cols [16w,16w+16)
#define LDS_STRIDE 264                // 256 + 8: lane-halves of a WMMA B-load hit disjoint banks

typedef float v2f __attribute__((ext_vector_type(2)));
typedef float v8f __attribute__((ext_vector_type(8)));

#if __has_builtin(__builtin_amdgcn_global_load_async_to_lds_b128) && \
    __has_builtin(__builtin_amdgcn_s_wait_asynccnt)
#define USE_ASYNC_LDS 1
typedef int v4i __attribute__((vector_size(16)));
typedef __attribute__((address_space(1))) v4i* g_v4i_ptr;   // global
typedef __attribute__((address_space(3))) v4i* l_v4i_ptr;   // LDS
#else
#define USE_ASYNC_LDS 0
#endif

// D[m][n] (16 nodes x 16 feat cols) accumulated over k via V_WMMA_F32_16X16X4_F32:
//   A[m][r] = 0.1 * (m == r)  (scaled selection, built in registers)
//   B[r][n] = staged row r, feat col n0+n   (from LDS)
__global__ __launch_bounds__(MEAN_THREADS)
void mean_wmma_kernel(const float* __restrict__ feat,
                      const int*   __restrict__ neigh,   // [B, K]
                      const int*   __restrict__ perm,    // [N]
                      float*       __restrict__ out_to,  // [B, D]
                      float*       __restrict__ out_shuf)// [B, D]
{
    __shared__ float xs[TILE_B * LDS_STRIDE];
    __shared__ int   rows0[TILE_B * K_S];   // direct neighbor rows
    __shared__ int   rows1[TILE_B * K_S];   // permuted neighbor rows

    const int tid  = threadIdx.x;
    const int b0   = blockIdx.x * TILE_B;
    const int wave = tid >> 5;              // 0..15 == staged row m
    const int lane = tid & 31;
    const int m16  = lane & 15;             // WMMA: M index (A) / N index (B,C,D)
    const int h    = lane >> 4;             // lane half
    const int n0   = wave * 16;             // this wave's 16-wide feature tile

    if (tid < TILE_B * K_S) {
        int m = tid / K_S, k = tid % K_S;
        int idx = neigh[(size_t)(b0 + m) * K_S + k];
        rows0[m * K_S + k] = idx;
        rows1[m * K_S + k] = perm[idx];
    }
    __syncthreads();

    for (int pass = 0; pass < 2; ++pass) {
        const int* rows = pass ? rows1 : rows0;
        float*     outp = pass ? out_shuf : out_to;
        v8f acc = {};

        for (int k = 0; k < K_S; ++k) {
            __syncthreads();                        // previous readers done with xs
            {   // wave w gathers its row (1 KB) into LDS
                int r = rows[wave * K_S + k];
#if USE_ASYNC_LDS
                // CDNA5 async copy: memory -> LDS, no VGPR round-trip (ASYNCcnt).
                // INST_OFFSET is added to BOTH global and LDS addresses (ISA 08 §4.4),
                // so the second 512B half reuses the same address registers.
                const float* src = feat + (size_t)r * D_FEAT + lane * 4;
                float*       dst = xs + wave * LDS_STRIDE + lane * 4;
                __builtin_amdgcn_global_load_async_to_lds_b128(
                    (g_v4i_ptr)src, (l_v4i_ptr)dst, 0,   0);
                __builtin_amdgcn_global_load_async_to_lds_b128(
                    (g_v4i_ptr)src, (l_v4i_ptr)dst, 512, 0);
#else
                const float4* src = (const float4*)(feat + (size_t)r * D_FEAT);
                float4* dst = (float4*)(xs + wave * LDS_STRIDE);
                dst[lane]      = src[lane];
                dst[lane + 32] = src[lane + 32];
#endif
            }
            if (k + 1 < K_S) {                      // pull next gather row toward the WGP
                int rn = rows[wave * K_S + k + 1];
                __builtin_prefetch(feat + (size_t)rn * D_FEAT + lane * 8, 0, 3);
            }
#if USE_ASYNC_LDS
            __builtin_amdgcn_s_wait_asynccnt(0);    // our LDS writes have landed
#endif
            __syncthreads();

            #pragma unroll
            for (int c = 0; c < 4; ++c) {           // K=16 staged rows, 4 per WMMA
                const int r0 = 4 * c + 2 * h;       // row covered by VGPR j=0 in this lane-half
                v2f a, b;
                a.x = (m16 == r0)     ? 0.1f : 0.0f;
                a.y = (m16 == r0 + 1) ? 0.1f : 0.0f;
                b.x = xs[ r0      * LDS_STRIDE + n0 + m16];
                b.y = xs[(r0 + 1) * LDS_STRIDE + n0 + m16];
                acc = __builtin_amdgcn_wmma_f32_16x16x4_f32(
                        /*neg_a=*/false, a, /*neg_b=*/false, b,
                        /*c_mod=*/(short)0, acc, /*reuse_a=*/false, /*reuse_b=*/false);
            }
        }

        // C/D layout: VGPR i, lane L -> M = i + 8*(L>>4), N = L&15
        #pragma unroll
        for (int i = 0; i < 8; ++i) {
            int m = i + 8 * h;
            outp[(size_t)(b0 + m) * D_FEAT + n0 + m16] = acc[i];
        }
    }
}

// Pure row gathers: out2[b] = feat[nodes[b]], out3[b] = feat[perm[nodes[b]]].
__global__ __launch_bounds__(256)
void skip_gather_kernel(const float* __restrict__ feat,
                        const int*   __restrict__ nodes,
                        const int*   __restrict__ perm,
                        float*       __restrict__ out_skip,
                        float*       __restrict__ out_sskip,
                        int total4)                     // batch * D/4
{
    int gid = blockIdx.x * blockDim.x + threadIdx.x;
    if (gid >= total4) return;
    int row = gid >> 6;                                 // D/4 == 64 float4 per row
    int c4  = gid & 63;
    int n   = nodes[row];
    ((float4*)(out_skip  + (size_t)row * D_FEAT))[c4] =
        ((const float4*)(feat + (size_t)n * D_FEAT))[c4];
    int p   = perm[n];
    ((float4*)(out_sskip + (size_t)row * D_FEAT))[c4] =
        ((const float4*)(feat + (size_t)p * D_FEAT))[c4];
}

extern "C" void kernel_launch(void* const* d_in, const int* in_sizes, int n_in,
                              void* d_out, int out_size, void* d_ws, size_t ws_size,
                              hipStream_t stream)
{
    const float* feat  = (const float*)d_in[0];   // [N, 256] f32
    const int*   neigh = (const int*)  d_in[1];   // [B, 10]
    const int*   nodes = (const int*)  d_in[2];   // [B]
    const int*   perm  = (const int*)  d_in[3];   // [N]

    const int batch = in_sizes[2];                // 16384
    float* out       = (float*)d_out;
    float* out_to    = out;
    float* out_shuf  = out + 1 * (size_t)batch * D_FEAT;
    float* out_skip  = out + 2 * (size_t)batch * D_FEAT;
    float* out_sskip = out + 3 * (size_t)batch * D_FEAT;

    mean_wmma_kernel<<<batch / TILE_B, MEAN_THREADS, 0, stream>>>(
        feat, neigh, perm, out_to, out_shuf);

    int total4 = batch * (D_FEAT / 4);
    skip_gather_kernel<<<(total4 + 255) / 256, 256, 0, stream>>>(
        feat, nodes, perm, out_skip, out_sskip, total4);
}